// SGLayerEpilogue_40321152974936
// MI455X (gfx1250) — compile-verified
//
#include <hip/hip_runtime.h>

// ---------------------------------------------------------------------------
// StyleGAN layer epilogue, MI455X (gfx1250, wave32).
//   x[16,512,64,64] -> noise-add -> LeakyReLU(0.2) -> PixelNorm(C) ->
//   InstanceNorm(HW) -> StyleMod(style @ lin_W + lin_b)
//
// 3-kernel plan (bandwidth-bound; see analysis):
//   k_style_gemm : 16x1024x512 fp32 GEMM via V_WMMA_F32_16X16X4_F32
//   k_pixel_stats: stream x once, produce r[b,hw] = rsqrt(mean_c y^2 + 1e-8)
//   k_inst_style : per-(b,c) block; recompute z, instance stats, fused output
// ---------------------------------------------------------------------------

typedef __attribute__((ext_vector_type(2))) float v2f;
typedef __attribute__((ext_vector_type(8))) float v8f;

#define B_     16
#define C_     512
#define HW_    4096      // 64*64
#define S_     512
#define TWO_C  1024

#define EPS_PIXEL 1e-8f
#define EPS_INST  1e-5f
#define NEG_SLOPE 0.2f

// ---------------------------------------------------------------------------
// Kernel 0: s[16,1024] = style[16,512] @ lin_W[512,1024] + lin_b[1024]
// One wave (32 lanes) per 16-wide N tile; K stepped by 4 with fp32 WMMA.
//
// ISA 7.12.2 layouts (wave32):
//   A 16x4 f32 : lanes 0-15 -> M=lane (K=0,1 in v0,v1); lanes 16-31 -> K=2,3
//   B 4x16 f32 : v0 lanes0-15 K=0, lanes16-31 K=2; v1 -> K=1 / K=3; N=lane&15
//   C/D 16x16  : vgpr r, lanes0-15 -> M=r, lanes16-31 -> M=r+8; N=lane&15
// ---------------------------------------------------------------------------
__global__ __launch_bounds__(32) void k_style_gemm(
    const float* __restrict__ style,   // [16, 512]
    const float* __restrict__ linW,    // [512, 1024]
    const float* __restrict__ linb,    // [1024]
    float* __restrict__ s_out)         // [16, 1024]
{
    const int lane = threadIdx.x;          // 0..31
    const int ncol = blockIdx.x << 4;      // N tile base (64 tiles)
    const int mn   = lane & 15;            // M for A, N for B/D
    const int kh   = (lane >> 4) << 1;     // 0 (lanes 0-15) or 2 (lanes 16-31)

    v8f acc = {};
#pragma unroll 4
    for (int k0 = 0; k0 < S_; k0 += 4) {
        v2f a, b;
        a.x = style[mn * S_ + k0 + kh];
        a.y = style[mn * S_ + k0 + kh + 1];
        b.x = linW[(size_t)(k0 + kh)     * TWO_C + ncol + mn];
        b.y = linW[(size_t)(k0 + kh + 1) * TWO_C + ncol + mn];
        // 8 args: (neg_a, A, neg_b, B, c_mod, C, reuse_a, reuse_b)
        acc = __builtin_amdgcn_wmma_f32_16x16x4_f32(
            false, a, false, b, (short)0, acc, false, false);
    }

    const float bias  = linb[ncol + mn];
    const int   mbase = (lane >> 4) << 3;  // 0 or 8
#pragma unroll
    for (int r = 0; r < 8; ++r)
        s_out[(size_t)(mbase + r) * TWO_C + ncol + mn] = acc[r] + bias;
}

// ---------------------------------------------------------------------------
// Kernel 1: PixelNorm scale.
//   r[b,hw] = rsqrt( mean_c( leaky(x + nw[c]*noise)^2 ) + 1e-8 )
// Block = 256 threads = 256 consecutive pixels of one batch; loop over C.
// Consecutive threads read consecutive addresses per channel (coalesced b32).
// ---------------------------------------------------------------------------
__global__ __launch_bounds__(256) void k_pixel_stats(
    const float* __restrict__ x,       // [16, 512, 4096]
    const float* __restrict__ noise,   // [16, 4096]
    const float* __restrict__ nw,      // [512]
    float* __restrict__ rinv)          // [16, 4096]
{
    __shared__ float snw[C_];
    // stage noise_weight in LDS (2 KB) once per block
    for (int i = threadIdx.x; i < C_; i += 256) snw[i] = nw[i];
    __syncthreads();

    const int blk = blockIdx.x;              // 0..255
    const int b   = blk >> 4;                // 16 pixel-chunks per batch
    const int pix = ((blk & 15) << 8) + threadIdx.x;

    const float* xb = x + (size_t)b * C_ * HW_ + pix;
    const float  nz = noise[(size_t)b * HW_ + pix];

    float acc = 0.0f;
    for (int c = 0; c < C_; c += 8) {
        if (c + 8 < C_)  // pull the next channel block toward L2/L0
            __builtin_prefetch(xb + (size_t)(c + 8) * HW_, 0, 3);
#pragma unroll
        for (int u = 0; u < 8; ++u) {
            float y = xb[(size_t)(c + u) * HW_] + snw[c + u] * nz;
            y = (y >= 0.0f) ? y : NEG_SLOPE * y;
            acc = fmaf(y, y, acc);
        }
    }
    rinv[(size_t)b * HW_ + pix] = rsqrtf(acc * (1.0f / C_) + EPS_PIXEL);
}

// ---------------------------------------------------------------------------
// Kernel 2: InstanceNorm + StyleMod, one block per (b,c).
// 256 threads x 16 floats = 4096-element channel tile held in registers
// (b128 loads). noise[b]/rinv[b] (16 KB each) are reused by 512 blocks -> L2.
// out = z*scale + shift with scale = rstd*(s0+1), shift = s1 - mean*scale.
// ---------------------------------------------------------------------------
__global__ __launch_bounds__(256) void k_inst_style(
    const float* __restrict__ x,       // [16, 512, 4096]
    const float* __restrict__ noise,   // [16, 4096]
    const float* __restrict__ nw,      // [512]
    const float* __restrict__ rinv,    // [16, 4096]
    const float* __restrict__ s,       // [16, 1024]
    float* __restrict__ out)           // [16, 512, 4096]
{
    const int bc = blockIdx.x;         // 0..8191
    const int b  = bc >> 9;
    const int c  = bc & (C_ - 1);

    const size_t base = (size_t)bc * HW_;
    const float4* xv = (const float4*)(x + base);
    const float4* nv = (const float4*)(noise + (size_t)b * HW_);
    const float4* rv = (const float4*)(rinv  + (size_t)b * HW_);

    const float w = nw[c];

    float4 z[4];
    float sum = 0.0f, ssq = 0.0f;
#pragma unroll
    for (int i = 0; i < 4; ++i) {
        const int idx = (i << 8) + threadIdx.x;   // 0..1023 float4s
        const float4 xx = xv[idx];
        const float4 nn = nv[idx];
        const float4 rr = rv[idx];
        float4 zz;
        zz.x = fmaf(w, nn.x, xx.x); zz.x = (zz.x >= 0.f) ? zz.x : NEG_SLOPE * zz.x; zz.x *= rr.x;
        zz.y = fmaf(w, nn.y, xx.y); zz.y = (zz.y >= 0.f) ? zz.y : NEG_SLOPE * zz.y; zz.y *= rr.y;
        zz.z = fmaf(w, nn.z, xx.z); zz.z = (zz.z >= 0.f) ? zz.z : NEG_SLOPE * zz.z; zz.z *= rr.z;
        zz.w = fmaf(w, nn.w, xx.w); zz.w = (zz.w >= 0.f) ? zz.w : NEG_SLOPE * zz.w; zz.w *= rr.w;
        z[i] = zz;
        sum += zz.x + zz.y + zz.z + zz.w;
        ssq  = fmaf(zz.x, zz.x, ssq); ssq = fmaf(zz.y, zz.y, ssq);
        ssq  = fmaf(zz.z, zz.z, ssq); ssq = fmaf(zz.w, zz.w, ssq);
    }

    // wave32 butterfly reduction, then cross-wave via LDS (8 waves)
#pragma unroll
    for (int off = 16; off > 0; off >>= 1) {
        sum += __shfl_xor(sum, off, 32);
        ssq += __shfl_xor(ssq, off, 32);
    }
    __shared__ float lsum[8], lssq[8];
    const int wave = threadIdx.x >> 5;
    if ((threadIdx.x & 31) == 0) { lsum[wave] = sum; lssq[wave] = ssq; }
    __syncthreads();
    float tsum = 0.0f, tssq = 0.0f;
#pragma unroll
    for (int wv = 0; wv < 8; ++wv) { tsum += lsum[wv]; tssq += lssq[wv]; }

    const float mean = tsum * (1.0f / HW_);
    const float var  = tssq * (1.0f / HW_) - mean * mean;
    const float rstd = rsqrtf(var + EPS_INST);

    const float s0    = s[(size_t)b * TWO_C + c] + 1.0f;
    const float s1    = s[(size_t)b * TWO_C + C_ + c];
    const float scale = rstd * s0;
    const float shift = fmaf(-mean, scale, s1);

    float4* ov = (float4*)(out + base);
#pragma unroll
    for (int i = 0; i < 4; ++i) {
        const int idx = (i << 8) + threadIdx.x;
        float4 oo;
        oo.x = fmaf(z[i].x, scale, shift);
        oo.y = fmaf(z[i].y, scale, shift);
        oo.z = fmaf(z[i].z, scale, shift);
        oo.w = fmaf(z[i].w, scale, shift);
        ov[idx] = oo;
    }
}

// ---------------------------------------------------------------------------
extern "C" void kernel_launch(void* const* d_in, const int* in_sizes, int n_in,
                              void* d_out, int out_size, void* d_ws, size_t ws_size,
                              hipStream_t stream) {
    // setup_inputs() order: x, noise, style, noise_weight, lin_W, lin_b (all fp32)
    const float* x     = (const float*)d_in[0];
    const float* noise = (const float*)d_in[1];
    const float* style = (const float*)d_in[2];
    const float* nw    = (const float*)d_in[3];
    const float* linW  = (const float*)d_in[4];
    const float* linb  = (const float*)d_in[5];
    float*       out   = (float*)d_out;

    // workspace: s[16,1024] (64 KB) then rinv[16,4096] (256 KB); both fully
    // rewritten every call, so 0xAA poison is harmless.
    float* s_buf = (float*)d_ws;
    float* rinv  = s_buf + (size_t)B_ * TWO_C;

    k_style_gemm <<<TWO_C / 16,        32,  0, stream>>>(style, linW, linb, s_buf);
    k_pixel_stats<<<B_ * (HW_ / 256),  256, 0, stream>>>(x, noise, nw, rinv);
    k_inst_style <<<B_ * C_,           256, 0, stream>>>(x, noise, nw, rinv, s_buf, out);
}